// RNNTLoss_70927089926543
// MI455X (gfx1250) — compile-verified
//
#include <hip/hip_runtime.h>
#include <cstdint>
#include <cstddef>

// ---------------------------------------------------------------------------
// RNNT loss (forward) for MI455X / gfx1250.
//   Phase 1: streaming log-softmax over V=1024, extract blank+label logprobs.
//            Bandwidth bound: 272.6 MB @ 23.3 TB/s ~= 11.7 us floor.
//            CDNA5 async global->LDS loads (ASYNCcnt) with a 2-row double
//            buffer per wave, NT temporal hint (read-once stream, 272 MB >
//            192 MB L2 -> don't thrash L2; keeps lpb/lpl L2-resident for the
//            DP kernel). Wave32 shuffle reductions.
//   Phase 2: anti-diagonal wavefront DP over the (T, U+1) lattice, one
//            workgroup per batch element, alpha kept in LDS (33 KB << 320 KB).
//   Phase 3: sum per-batch costs -> out[0].
// No matmul structure exists in RNNT loss, so WMMA is not applicable; the
// gfx1250-specific path exercised is GLOBAL_LOAD_ASYNC_TO_LDS_B128 +
// S_WAIT_ASYNCCNT.
// ---------------------------------------------------------------------------

#define NEGV (-1e30f)

constexpr int V_DIM    = 1024;  // vocab size (reference), 32 floats per lane
constexpr int WAVE     = 32;    // gfx1250 is wave32
constexpr int K1_WAVES = 4;     // waves per block in kernel 1 (32 KB LDS)
constexpr int K1_RPW   = 4;     // rows per wave (amortizes pipeline fill)

// LDS byte offset of a pointer into a __shared__ array (addrspace(3) ptrs are
// 32-bit LDS offsets; HW adds LDS_BASE to the async op's VDST value).
__device__ __forceinline__ unsigned lds_off(const float* p) {
  return (unsigned)(size_t)(__attribute__((address_space(3))) const float*)p;
}

// One 4 KB row: 8 issues x (32 lanes x 16 B) fully coalesced async copies.
// NT hint: acts is a read-once 272 MB stream; do not displace L2 contents.
__device__ __forceinline__ void async_row_load(const float* g, float* l,
                                               int lane) {
#pragma unroll
  for (int c = 0; c < 8; ++c) {
    const float* gp = g + c * 128 + lane * 4;
    unsigned lo = lds_off(l + c * 128 + lane * 4);
    asm volatile("global_load_async_to_lds_b128 %0, %1, off th:TH_LOAD_NT"
                 :: "v"(lo), "v"(gp)
                 : "memory");
  }
}

template <int N>
__device__ __forceinline__ void wait_asynccnt() {
  asm volatile("s_wait_asynccnt %0" :: "i"(N) : "memory");
}

// ---------------------------------------------------------------------------
// Kernel 1: per-row log-softmax, keep lp_blank (B,T,U+1) and lp_label (B,T,U).
// One wave per (b,t,u) row of V=1024 logits.
// ---------------------------------------------------------------------------
__global__ __launch_bounds__(K1_WAVES * WAVE) void
rnnt_lse_extract(const float* __restrict__ acts,
                 const int* __restrict__ labels,      // (B,U)
                 const int* __restrict__ label_lens,  // (B)
                 float* __restrict__ lpb,             // (B,T,U+1)
                 float* __restrict__ lpl,             // (B,T,U)
                 int nrows, int T, int U) {
  __shared__ __align__(16) float sbuf[K1_WAVES * 2 * V_DIM];
  const int wid  = threadIdx.x >> 5;
  const int lane = threadIdx.x & 31;
  float* bufA = &sbuf[(wid * 2 + 0) * V_DIM];
  float* bufB = &sbuf[(wid * 2 + 1) * V_DIM];

  const int U1 = U + 1;
  const int gw = blockIdx.x * K1_WAVES + wid;  // global wave id
  const int nw = gridDim.x * K1_WAVES;         // total waves

  int row = gw;
  if (row < nrows) async_row_load(acts + (size_t)row * V_DIM, bufA, lane);
  int cur = 0;
  for (; row < nrows; row += nw) {
    const int nxt = row + nw;
    float* lb = cur ? bufB : bufA;
    float* nb = cur ? bufA : bufB;
    if (nxt < nrows) {
      async_row_load(acts + (size_t)nxt * V_DIM, nb, lane);
      wait_asynccnt<8>();  // current row's 8 copies done; prefetch in flight
    } else {
      wait_asynccnt<0>();
    }

    // 32 elements per lane from LDS (ds_load_b128 granularity).
    float4 v[8];
#pragma unroll
    for (int c = 0; c < 8; ++c)
      v[c] = *(const float4*)&lb[c * 128 + lane * 4];

    float m = v[0].x;
#pragma unroll
    for (int c = 0; c < 8; ++c) {
      m = fmaxf(m, v[c].x); m = fmaxf(m, v[c].y);
      m = fmaxf(m, v[c].z); m = fmaxf(m, v[c].w);
    }
#pragma unroll
    for (int d = 16; d >= 1; d >>= 1) m = fmaxf(m, __shfl_xor(m, d, WAVE));

    float s = 0.f;
#pragma unroll
    for (int c = 0; c < 8; ++c) {
      s += __expf(v[c].x - m) + __expf(v[c].y - m) +
           __expf(v[c].z - m) + __expf(v[c].w - m);
    }
#pragma unroll
    for (int d = 16; d >= 1; d >>= 1) s += __shfl_xor(s, d, WAVE);

    const float lse = m + __logf(s);

    if (lane == 0) {
      const int u  = row % U1;
      const int bt = row / U1;   // b*T + t
      const int b  = bt / T;
      lpb[row] = lb[0] - lse;    // blank = index 0
      if (u < U) {
        const int lab = labels[b * U + u];
        const float val = (u < label_lens[b]) ? (lb[lab] - lse) : NEGV;
        lpl[bt * U + u] = val;
      }
    }
    cur ^= 1;
  }
}

// ---------------------------------------------------------------------------
// Kernel 2: forward DP over the lattice, anti-diagonal wavefront.
// One block per batch element; alpha (T x U+1) lives in dynamic LDS.
// ---------------------------------------------------------------------------
__global__ __launch_bounds__(128) void
rnnt_dp(const float* __restrict__ lpb, const float* __restrict__ lpl,
        const int* __restrict__ act_lens, const int* __restrict__ label_lens,
        float* __restrict__ costs, int T, int U) {
  extern __shared__ float alpha[];  // T*(U+1) floats
  const int b  = blockIdx.x;
  const int U1 = U + 1;
  const float* pb = lpb + (size_t)b * T * U1;
  const float* pl = lpl + (size_t)b * T * U;

  for (int d = 0; d < T + U; ++d) {
    const int tmin = max(0, d - U);
    const int tmax = min(T - 1, d);
    const int t = tmin + (int)threadIdx.x;
    if (t <= tmax) {
      const int u = d - t;
      float a;
      if (d == 0) {
        a = 0.f;  // alpha[0,0]
      } else {
        const float va = (t > 0) ? alpha[(t - 1) * U1 + u] + pb[(t - 1) * U1 + u]
                                 : NEGV;
        const float vb = (u > 0) ? alpha[t * U1 + (u - 1)] + pl[t * U + (u - 1)]
                                 : NEGV;
        const float hi = fmaxf(va, vb);
        const float lo = fminf(va, vb);
        // logaddexp; exp underflows to 0 when one side is NEGV -> returns hi
        a = hi + __logf(1.f + __expf(lo - hi));
      }
      alpha[t * U1 + u] = a;
    }
    __syncthreads();
  }

  if (threadIdx.x == 0) {
    const int tl = act_lens[b] - 1;
    const int ul = label_lens[b];
    costs[b] = -(alpha[tl * U1 + ul] + pb[tl * U1 + ul]);
  }
}

// ---------------------------------------------------------------------------
// Kernel 3: out[0] = sum_b costs[b]
// ---------------------------------------------------------------------------
__global__ void rnnt_sum(const float* __restrict__ costs,
                         float* __restrict__ out, int B) {
  if (threadIdx.x == 0) {
    float s = 0.f;
    for (int i = 0; i < B; ++i) s += costs[i];
    out[0] = s;
  }
}

// ---------------------------------------------------------------------------
extern "C" void kernel_launch(void* const* d_in, const int* in_sizes, int n_in,
                              void* d_out, int out_size, void* d_ws,
                              size_t ws_size, hipStream_t stream) {
  const float* acts       = (const float*)d_in[0];
  const int*   labels     = (const int*)d_in[1];
  const int*   act_lens   = (const int*)d_in[2];
  const int*   label_lens = (const int*)d_in[3];
  float*       out        = (float*)d_out;

  const int B  = in_sizes[2];
  const int U  = in_sizes[1] / B;
  const int U1 = U + 1;
  const int T  = (int)((long long)in_sizes[0] / ((long long)B * U1 * V_DIM));
  const int nrows = B * T * U1;  // 66,560 for the reference shapes

  // Workspace: lpb (nrows) | lpl (B*T*U) | costs (B)   ~= 529 KB total
  float* lpb   = (float*)d_ws;
  float* lpl   = lpb + nrows;
  float* costs = lpl + (size_t)B * T * U;

  const int rows_per_block = K1_WAVES * K1_RPW;
  const int grid1 = (nrows + rows_per_block - 1) / rows_per_block;
  rnnt_lse_extract<<<grid1, K1_WAVES * WAVE, 0, stream>>>(
      acts, labels, label_lens, lpb, lpl, nrows, T, U);

  rnnt_dp<<<B, 128, (size_t)T * U1 * sizeof(float), stream>>>(
      lpb, lpl, act_lens, label_lens, costs, T, U);

  rnnt_sum<<<1, WAVE, 0, stream>>>(costs, out, B);
}